// Colorizer_48567490183659
// MI455X (gfx1250) — compile-verified
//
#include <hip/hip_runtime.h>
#include <math.h>

#define PI_F 3.14159265358979323846f

typedef __attribute__((ext_vector_type(2))) float v2f;
typedef __attribute__((ext_vector_type(8))) float v8f;

static inline int ceil_div_i(long long a, long long b) { return (int)((a + b - 1) / b); }

// ---------------------------------------------------------------------------
// Weight transform: OIHW -> [kpos][cb4][khalf][coutP][pair], where the K value
// held is c = cb4*4 + 2*khalf + pair. This makes each lane's A-fragment one
// contiguous 8-byte load, coalesced across lanes (stride 8B over m).
// Zero padding for m >= COUT.
// ---------------------------------------------------------------------------
template <int CIN, int COUT, int COUTP, int KS>
__global__ void wt_kernel(const float* __restrict__ w, float* __restrict__ wT) {
    int i = blockIdx.x * blockDim.x + threadIdx.x;
    constexpr int TOTAL = KS * KS * CIN * COUTP;
    if (i >= TOTAL) return;
    int pr = i & 1;
    int rest = i >> 1;
    int m = rest % COUTP;
    int rest2 = rest / COUTP;
    int kh = rest2 & 1;
    int rest3 = rest2 >> 1;
    int cb4 = rest3 % (CIN / 4);
    int kpos = rest3 / (CIN / 4);
    int ky = kpos / KS, kx = kpos % KS;
    int c = cb4 * 4 + 2 * kh + pr;
    float v = 0.f;
    if (m < COUT) v = w[(((size_t)m * CIN + c) * KS + ky) * KS + kx];
    wT[i] = v;
}

// ---------------------------------------------------------------------------
// Dense SIFT stage 1: orientation-binned gradient magnitude, 4x4 pooled.
// Output "pooled": (B, H+1, W+1, 8) channel-last.
// ---------------------------------------------------------------------------
__global__ void sift_pool_kernel(const float* __restrict__ x, float* __restrict__ pooled,
                                 int B, int H, int W) {
    int HP = H + 1, WP = W + 1;
    size_t i = (size_t)blockIdx.x * blockDim.x + threadIdx.x;
    size_t total = (size_t)B * HP * WP;
    if (i >= total) return;
    int X = (int)(i % WP);
    int Y = (int)((i / WP) % HP);
    int b = (int)(i / ((size_t)WP * HP));
    const float* xb = x + (size_t)b * H * W;

    float acc[8];
#pragma unroll
    for (int a = 0; a < 8; ++a) acc[a] = 0.f;
    const float xc2[4] = {0.5f, 1.5f, 1.5f, 0.5f};

#pragma unroll
    for (int di = 0; di < 4; ++di) {
#pragma unroll
        for (int dj = 0; dj < 4; ++dj) {
            int yy = Y + di - 2, xx = X + dj - 2;
            if (yy < 0 || yy >= H || xx < 0 || xx >= W) continue;
            int xm = xx - 1 < 0 ? 0 : xx - 1;
            int xp = xx + 1 >= W ? W - 1 : xx + 1;
            int ym = yy - 1 < 0 ? 0 : yy - 1;
            int yp = yy + 1 >= H ? H - 1 : yy + 1;
            float gx = (xb[(size_t)yy * W + xp] - xb[(size_t)yy * W + xm]) * 0.5f;
            float gy = (xb[(size_t)yp * W + xx] - xb[(size_t)ym * W + xx]) * 0.5f;
            float mag = sqrtf(gx * gx + gy * gy + 1e-10f);
            float ori = atan2f(gy, gx + 1e-10f) + 2.0f * PI_F;
            float obig = 8.0f * ori / (2.0f * PI_F);
            float b0f = floorf(obig);
            float w1f = obig - b0f;
            float w0 = (1.0f - w1f) * mag;
            float w1 = w1f * mag;
            int bo0 = ((int)b0f) & 7;   // b0f >= 0 here, so &7 == %8
            int bo1 = (bo0 + 1) & 7;
            float pw = xc2[di] * xc2[dj] * 0.25f;
#pragma unroll
            for (int a = 0; a < 8; ++a)
                acc[a] += (a == bo0 ? pw * w0 : 0.f) + (a == bo1 ? pw * w1 : 0.f);
        }
    }
    float* o = pooled + i * 8;
#pragma unroll
    for (int a = 0; a < 8; ++a) o[a] = acc[a];
}

// ---------------------------------------------------------------------------
// Dense SIFT stage 2: 4x4x8 gather + L2 / clip(0.2) / L2 / L1 / sqrt.
// Output f0: (B, H, W, 128) channel-last, c = a*16 + ii*4 + jj.
// ---------------------------------------------------------------------------
__global__ void sift_desc_kernel(const float* __restrict__ pooled, float* __restrict__ f0,
                                 int B, int H, int W) {
    int HP = H + 1, WP = W + 1;
    size_t i = (size_t)blockIdx.x * blockDim.x + threadIdx.x;
    size_t total = (size_t)B * H * W;
    if (i >= total) return;
    int xx = (int)(i % W);
    int yy = (int)((i / W) % H);
    int b = (int)(i / ((size_t)H * W));
    const float* pb = pooled + (size_t)b * HP * WP * 8;

    auto ld = [&](int pos, int a) -> float {
        int ii = pos >> 2, jj = pos & 3;
        int py = yy + ii - 1, px = xx + jj - 1;
        if (py < 0 || py >= HP || px < 0 || px >= WP) return 0.f;
        return pb[((size_t)py * WP + px) * 8 + a];
    };

    float sumsq = 0.f;
    for (int pos = 0; pos < 16; ++pos)
#pragma unroll
        for (int a = 0; a < 8; ++a) { float v = ld(pos, a); sumsq += v * v; }
    float inv1 = 1.0f / fmaxf(sqrtf(sumsq), 1e-12f);

    float sumsq2 = 0.f, sumw = 0.f;
    for (int pos = 0; pos < 16; ++pos)
#pragma unroll
        for (int a = 0; a < 8; ++a) {
            float wv = fminf(fmaxf(ld(pos, a) * inv1, 0.f), 0.2f);
            sumsq2 += wv * wv;
            sumw += wv;
        }
    float inv2 = 1.0f / fmaxf(sqrtf(sumsq2), 1e-12f);
    float invl1 = 1.0f / fmaxf(sumw * inv2, 1e-12f);

    float* o = f0 + i * 128;
    for (int pos = 0; pos < 16; ++pos)
#pragma unroll
        for (int a = 0; a < 8; ++a) {
            float wv = fminf(fmaxf(ld(pos, a) * inv1, 0.f), 0.2f);
            o[a * 16 + pos] = sqrtf(wv * inv2 * invl1 + 1e-10f);
        }
}

// ---------------------------------------------------------------------------
// Implicit-GEMM conv via V_WMMA_F32_16X16X4_F32. Channel-last tensors.
// One block = 16 consecutive pixels (one row segment) x all out channels.
// One wave = one 16-row M tile. LDS stages the KS x (16+2P) x CIN halo tile
// with per-pixel stride CIN+4 (bank-conflict-free: 16 lanes land 4 banks
// apart) and channels bit-reverse permuted within groups of 4 so each lane's
// B-fragment (K rows cb+khalf, cb+khalf+2) is one contiguous ds_load_b64.
// A-fragments are one coalesced global_load_b64 from the transformed weights.
// ---------------------------------------------------------------------------
template <int CIN, int COUT, int COUTP, int KS, int PAD>
__global__ __launch_bounds__((COUTP / 16) * 32)
void conv_wmma_kernel(const float* __restrict__ in, const float* __restrict__ wT,
                      const float* __restrict__ bias, float* __restrict__ out,
                      int H, int W) {
    constexpr int NW = COUTP / 16;
    constexpr int COLS = 16 + 2 * PAD;
    constexpr int CINP = CIN + 4;  // padded LDS stride per pixel
    extern __shared__ float lds[];  // KS * COLS * CINP floats

    const int tpr = W / 16;
    const int tile = blockIdx.x;
    const int x0 = (tile % tpr) * 16;
    const int y = (tile / tpr) % H;
    const int b = tile / (tpr * H);
    const int tid = threadIdx.x;
    const int lane = tid & 31;
    const int wave = tid >> 5;

    // cooperatively stage halo tile (zero padded), channel groups of 4 stored
    // in order {0,2,1,3} so khalf reads stored offsets {2k, 2k+1}.
    constexpr int E = KS * COLS * CIN;
    for (int i = tid; i < E; i += NW * 32) {
        int c = i % CIN;
        int rest = i / CIN;
        int cc = rest % COLS;
        int r = rest / COLS;
        int yy = y - PAD + r;
        int xx = x0 - PAD + cc;
        float v = 0.f;
        if (yy >= 0 && yy < H && xx >= 0 && xx < W)
            v = in[((size_t)(b * H + yy) * W + xx) * CIN + c];
        int rbit = c & 3;
        int sc = (c & ~3) | ((rbit & 1) << 1) | (rbit >> 1);  // 0,2,1,3
        lds[(r * COLS + cc) * CINP + sc] = v;
    }
    __syncthreads();

    const int mrow = lane & 15;   // A-matrix row / B-matrix column
    const int khalf = lane >> 4;  // half-wave selector

    // C/D layout: VGPR r, lanes0-15 -> M=r, lanes16-31 -> M=r+8
    v8f acc;
#pragma unroll
    for (int r = 0; r < 8; ++r) {
        int m = wave * 16 + r + khalf * 8;
        acc[r] = (m < COUT) ? bias[m] : 0.f;
    }

    for (int kpos = 0; kpos < KS * KS; ++kpos) {
        int ky = kpos / KS, kx = kpos % KS;
        const v2f* ldsrow =
            (const v2f*)&lds[((size_t)ky * COLS + (mrow + kx)) * CINP + 2 * khalf];
        const v2f* wrow =
            (const v2f*)&wT[(((size_t)kpos * (CIN / 4) * 2 + khalf) * COUTP +
                             wave * 16 + mrow) * 2];
#pragma unroll 4
        for (int cb4 = 0; cb4 < CIN / 4; ++cb4) {
            // B 4x16: v0 = rows K0|K1, v1 = rows K2|K3, N = lane&15
            v2f bfrag = ldsrow[cb4 * 2];             // 8B: stored {2kh,2kh+1}
            // A 16x4: v0 = K0|K2, v1 = K1|K3, M = lane&15
            v2f afrag = wrow[(size_t)cb4 * COUTP];   // 8B: K pair for this half
            acc = __builtin_amdgcn_wmma_f32_16x16x4_f32(
                false, afrag, false, bfrag, (short)0, acc, false, false);
        }
    }

#pragma unroll
    for (int r = 0; r < 8; ++r) {
        int m = wave * 16 + r + khalf * 8;
        if (m < COUT)
            out[((size_t)(b * H + y) * W + x0 + mrow) * COUT + m] = acc[r];
    }
}

// ---------------------------------------------------------------------------
// GroupNorm: reduction (one block per (b,g)) + fused apply/ReLU.
// ---------------------------------------------------------------------------
template <int C, int G>
__global__ void gn_reduce_kernel(const float* __restrict__ t, float* __restrict__ stats,
                                 int H, int W) {
    constexpr int Cg = C / G;
    int bg = blockIdx.x;
    int g = bg % G;
    int b = bg / G;
    size_t n = (size_t)Cg * H * W;
    double s = 0.0, ss = 0.0;
    for (size_t i = threadIdx.x; i < n; i += blockDim.x) {
        int cc = (int)(i % Cg);
        size_t p = i / Cg;
        int xx = (int)(p % W);
        int yy = (int)(p / W);
        float v = t[((size_t)(b * H + yy) * W + xx) * C + g * Cg + cc];
        s += v;
        ss += (double)v * v;
    }
    __shared__ double sh[512];
    sh[threadIdx.x] = s;
    sh[threadIdx.x + 256] = ss;
    __syncthreads();
    for (int off = 128; off > 0; off >>= 1) {
        if ((int)threadIdx.x < off) {
            sh[threadIdx.x] += sh[threadIdx.x + off];
            sh[threadIdx.x + 256] += sh[threadIdx.x + 256 + off];
        }
        __syncthreads();
    }
    if (threadIdx.x == 0) {
        double mu = sh[0] / (double)n;
        double var = sh[256] / (double)n - mu * mu;
        stats[bg * 2] = (float)mu;
        stats[bg * 2 + 1] = (float)(1.0 / sqrt(var + 1e-5));
    }
}

template <int C, int G, bool RELU>
__global__ void gn_apply_kernel(float* __restrict__ t, const float* __restrict__ stats,
                                const float* __restrict__ gamma, const float* __restrict__ beta,
                                int B, int H, int W) {
    constexpr int Cg = C / G;
    size_t i = (size_t)blockIdx.x * blockDim.x + threadIdx.x;
    size_t N = (size_t)B * H * W * C;
    if (i >= N) return;
    int c = (int)(i % C);
    int b = (int)(i / ((size_t)H * W * C));
    int g = c / Cg;
    float mu = stats[(b * G + g) * 2];
    float rs = stats[(b * G + g) * 2 + 1];
    float v = (t[i] - mu) * rs * gamma[c] + beta[c];
    if (RELU) v = fmaxf(v, 0.f);
    t[i] = v;
}

// ---------------------------------------------------------------------------
// Guide (UV from gt RGB), joint bilateral, rescale min/max, compose.
// ---------------------------------------------------------------------------
__global__ void guide_kernel(const float* __restrict__ gt, float* __restrict__ guide,
                             int B, int H, int W) {
    size_t i = (size_t)blockIdx.x * blockDim.x + threadIdx.x;
    size_t hw = (size_t)H * W;
    if (i >= (size_t)B * hw) return;
    int b = (int)(i / hw);
    size_t p = i % hw;
    float r = gt[((size_t)b * 3 + 0) * hw + p];
    float g = gt[((size_t)b * 3 + 1) * hw + p];
    float bl = gt[((size_t)b * 3 + 2) * hw + p];
    guide[i * 2 + 0] = -0.147f * r - 0.289f * g + 0.436f * bl;
    guide[i * 2 + 1] = 0.615f * r - 0.515f * g - 0.1f * bl;
}

__global__ void bilateral_kernel(const float* __restrict__ f, const float* __restrict__ guide,
                                 float* __restrict__ out, int B, int H, int W) {
    size_t i = (size_t)blockIdx.x * blockDim.x + threadIdx.x;
    size_t hw = (size_t)H * W;
    if (i >= (size_t)B * hw) return;
    int xx = (int)(i % W);
    int yy = (int)((i / W) % H);
    int b = (int)(i / hw);

    float g1d[5];
    float s = 0.f;
#pragma unroll
    for (int k = 0; k < 5; ++k) {
        float d = (float)(k - 2);
        g1d[k] = expf(-d * d / (2.0f * 1.5f * 1.5f));
        s += g1d[k];
    }
#pragma unroll
    for (int k = 0; k < 5; ++k) g1d[k] /= s;
    const float inv = -0.5f / (0.1f * 0.1f);

    const float* gb = guide + (size_t)b * hw * 2;
    const float* fb = f + (size_t)b * hw * 2;
    float gc0 = gb[((size_t)yy * W + xx) * 2];
    float gc1 = gb[((size_t)yy * W + xx) * 2 + 1];
    float num0 = 0.f, num1 = 0.f, den = 0.f;
#pragma unroll
    for (int dy = 0; dy < 5; ++dy) {
        int py = yy + dy - 2;
        py = py < 0 ? -py : (py >= H ? 2 * H - 2 - py : py);
#pragma unroll
        for (int dx = 0; dx < 5; ++dx) {
            int px = xx + dx - 2;
            px = px < 0 ? -px : (px >= W ? 2 * W - 2 - px : px);
            size_t q = ((size_t)py * W + px) * 2;
            float d0 = fabsf(gb[q] - gc0) + fabsf(gb[q + 1] - gc1);
            float wgt = g1d[dy] * g1d[dx] * expf(inv * d0 * d0);
            num0 += wgt * fb[q];
            num1 += wgt * fb[q + 1];
            den += wgt;
        }
    }
    out[i * 2] = num0 / den;
    out[i * 2 + 1] = num1 / den;
}

__device__ inline unsigned enc_ord(float f) {
    unsigned u = __float_as_uint(f);
    return (u & 0x80000000u) ? ~u : (u | 0x80000000u);
}
__device__ inline float dec_ord(unsigned u) {
    unsigned b = (u & 0x80000000u) ? (u & 0x7fffffffu) : ~u;
    return __uint_as_float(b);
}

__global__ void mm_init_kernel(unsigned* __restrict__ mm) {
    mm[0] = 0xFFFFFFFFu;  // min ch0
    mm[1] = 0x00000000u;  // max ch0
    mm[2] = 0xFFFFFFFFu;  // min ch1
    mm[3] = 0x00000000u;  // max ch1
}

__global__ void mm_reduce_kernel(const float* __restrict__ fb, unsigned* __restrict__ mm,
                                 long long n) {
    unsigned mn0 = 0xFFFFFFFFu, mx0 = 0u, mn1 = 0xFFFFFFFFu, mx1 = 0u;
    for (long long i = (long long)blockIdx.x * blockDim.x + threadIdx.x; i < n;
         i += (long long)gridDim.x * blockDim.x) {
        unsigned e0 = enc_ord(fb[i * 2]);
        unsigned e1 = enc_ord(fb[i * 2 + 1]);
        mn0 = min(mn0, e0); mx0 = max(mx0, e0);
        mn1 = min(mn1, e1); mx1 = max(mx1, e1);
    }
    __shared__ unsigned sh[4][256];
    sh[0][threadIdx.x] = mn0; sh[1][threadIdx.x] = mx0;
    sh[2][threadIdx.x] = mn1; sh[3][threadIdx.x] = mx1;
    __syncthreads();
    for (int off = 128; off > 0; off >>= 1) {
        if ((int)threadIdx.x < off) {
            sh[0][threadIdx.x] = min(sh[0][threadIdx.x], sh[0][threadIdx.x + off]);
            sh[1][threadIdx.x] = max(sh[1][threadIdx.x], sh[1][threadIdx.x + off]);
            sh[2][threadIdx.x] = min(sh[2][threadIdx.x], sh[2][threadIdx.x + off]);
            sh[3][threadIdx.x] = max(sh[3][threadIdx.x], sh[3][threadIdx.x + off]);
        }
        __syncthreads();
    }
    if (threadIdx.x == 0) {
        atomicMin(&mm[0], sh[0][0]);
        atomicMax(&mm[1], sh[1][0]);
        atomicMin(&mm[2], sh[2][0]);
        atomicMax(&mm[3], sh[3][0]);
    }
}

__global__ void compose_kernel(const float* __restrict__ x, const float* __restrict__ fb,
                               const unsigned* __restrict__ mm, float* __restrict__ out,
                               int B, int H, int W) {
    size_t i = (size_t)blockIdx.x * blockDim.x + threadIdx.x;
    size_t hw = (size_t)H * W;
    if (i >= (size_t)B * hw) return;
    int b = (int)(i / hw);
    size_t p = i % hw;
    float mn0 = dec_ord(mm[0]), mx0 = dec_ord(mm[1]);
    float mn1 = dec_ord(mm[2]), mx1 = dec_ord(mm[3]);
    float u = fb[i * 2], v = fb[i * 2 + 1];
    float cu = fminf(fmaxf((u - mn0) / (mx0 - mn0 + 1e-6f), 0.f), 1.f);
    float cv = fminf(fmaxf((v - mn1) / (mx1 - mn1 + 1e-6f), 0.f), 1.f);
    float U = cu * (2.0f * 0.436f) - 0.436f;
    float V = cv * (2.0f * 0.615f) - 0.615f;
    float Y = x[i];
    out[((size_t)b * 3 + 0) * hw + p] = Y + 1.14f * V;
    out[((size_t)b * 3 + 1) * hw + p] = Y - 0.396f * U - 0.581f * V;
    out[((size_t)b * 3 + 2) * hw + p] = Y + 2.029f * U;
}

// ---------------------------------------------------------------------------
// Launch
// ---------------------------------------------------------------------------
extern "C" void kernel_launch(void* const* d_in, const int* in_sizes, int n_in,
                              void* d_out, int out_size, void* d_ws, size_t ws_size,
                              hipStream_t stream) {
    const float* x   = (const float*)d_in[0];
    const float* gt  = (const float*)d_in[1];
    const float* w1  = (const float*)d_in[2];
    const float* b1  = (const float*)d_in[3];
    const float* g1w = (const float*)d_in[4];
    const float* g1b = (const float*)d_in[5];
    const float* w2  = (const float*)d_in[6];
    const float* b2  = (const float*)d_in[7];
    const float* g2w = (const float*)d_in[8];
    const float* g2b = (const float*)d_in[9];
    const float* w3  = (const float*)d_in[10];
    const float* b3  = (const float*)d_in[11];
    const float* g3w = (const float*)d_in[12];
    const float* g3b = (const float*)d_in[13];
    const float* w4  = (const float*)d_in[14];
    const float* b4  = (const float*)d_in[15];
    const float* g4w = (const float*)d_in[16];
    const float* g4b = (const float*)d_in[17];
    const float* w5  = (const float*)d_in[18];
    const float* b5  = (const float*)d_in[19];

    const int B = 4, H = 270, W = 512;
    const int HP = H + 1, WP = W + 1;
    const size_t HW = (size_t)H * W;

    float* ws = (float*)d_ws;
    size_t o_pooled = 0;
    size_t sz_pooled = (size_t)B * HP * WP * 8;
    size_t o_f0 = o_pooled + sz_pooled;
    size_t sz_f0 = (size_t)B * HW * 128;
    size_t o_c1 = o_f0 + sz_f0;
    size_t sz_c1 = (size_t)B * HW * 64;
    // region reuse: after conv1, f0 is dead -> post-conv1 activations live there
    size_t o_c2 = o_f0;
    size_t o_c3 = o_c2 + (size_t)B * HW * 32;
    size_t o_c4 = o_c3 + (size_t)B * HW * 16;
    size_t o_c5 = o_c4 + (size_t)B * HW * 8;
    size_t o_guide = o_c5 + (size_t)B * HW * 2;
    size_t o_fb = o_guide + (size_t)B * HW * 2;
    // small buffers after c1
    size_t o_wt1 = o_c1 + sz_c1;           size_t sz_wt1 = 25 * 128 * 64;
    size_t o_wt2 = o_wt1 + sz_wt1;         size_t sz_wt2 = 9 * 64 * 32;
    size_t o_wt3 = o_wt2 + sz_wt2;         size_t sz_wt3 = 9 * 32 * 16;
    size_t o_wt4 = o_wt3 + sz_wt3;         size_t sz_wt4 = 9 * 16 * 16;
    size_t o_wt5 = o_wt4 + sz_wt4;         size_t sz_wt5 = 9 * 8 * 16;
    size_t o_st1 = o_wt5 + sz_wt5;
    size_t o_st2 = o_st1 + 64;
    size_t o_st3 = o_st2 + 64;
    size_t o_st4 = o_st3 + 64;
    size_t o_mm = o_st4 + 64;
    unsigned* mm = (unsigned*)(ws + o_mm);

    const int ntiles = B * H * (W / 16);
    const long long NPX = (long long)B * HW;

    // weight transforms
    wt_kernel<128, 64, 64, 5><<<ceil_div_i(25 * 128 * 64, 256), 256, 0, stream>>>(w1, ws + o_wt1);
    wt_kernel<64, 32, 32, 3><<<ceil_div_i(9 * 64 * 32, 256), 256, 0, stream>>>(w2, ws + o_wt2);
    wt_kernel<32, 16, 16, 3><<<ceil_div_i(9 * 32 * 16, 256), 256, 0, stream>>>(w3, ws + o_wt3);
    wt_kernel<16, 8, 16, 3><<<ceil_div_i(9 * 16 * 16, 256), 256, 0, stream>>>(w4, ws + o_wt4);
    wt_kernel<8, 2, 16, 3><<<ceil_div_i(9 * 8 * 16, 256), 256, 0, stream>>>(w5, ws + o_wt5);

    // dense SIFT
    sift_pool_kernel<<<ceil_div_i((long long)B * HP * WP, 256), 256, 0, stream>>>(
        x, ws + o_pooled, B, H, W);
    sift_desc_kernel<<<ceil_div_i(NPX, 256), 256, 0, stream>>>(ws + o_pooled, ws + o_f0, B, H, W);

    // conv1 128->64 5x5 + GN(8) + ReLU    (LDS: 5*20*(128+4)*4 = 52800 B)
    conv_wmma_kernel<128, 64, 64, 5, 2><<<ntiles, 128, 5 * 20 * 132 * 4, stream>>>(
        ws + o_f0, ws + o_wt1, b1, ws + o_c1, H, W);
    gn_reduce_kernel<64, 8><<<B * 8, 256, 0, stream>>>(ws + o_c1, ws + o_st1, H, W);
    gn_apply_kernel<64, 8, true><<<ceil_div_i(NPX * 64, 256), 256, 0, stream>>>(
        ws + o_c1, ws + o_st1, g1w, g1b, B, H, W);

    // conv2 64->32 3x3 + GN(8) + ReLU     (LDS: 3*18*68*4 = 14688 B)
    conv_wmma_kernel<64, 32, 32, 3, 1><<<ntiles, 64, 3 * 18 * 68 * 4, stream>>>(
        ws + o_c1, ws + o_wt2, b2, ws + o_c2, H, W);
    gn_reduce_kernel<32, 8><<<B * 8, 256, 0, stream>>>(ws + o_c2, ws + o_st2, H, W);
    gn_apply_kernel<32, 8, true><<<ceil_div_i(NPX * 32, 256), 256, 0, stream>>>(
        ws + o_c2, ws + o_st2, g2w, g2b, B, H, W);

    // conv3 32->16 3x3 + GN(4) + ReLU     (LDS: 3*18*36*4 = 7776 B)
    conv_wmma_kernel<32, 16, 16, 3, 1><<<ntiles, 32, 3 * 18 * 36 * 4, stream>>>(
        ws + o_c2, ws + o_wt3, b3, ws + o_c3, H, W);
    gn_reduce_kernel<16, 4><<<B * 4, 256, 0, stream>>>(ws + o_c3, ws + o_st3, H, W);
    gn_apply_kernel<16, 4, true><<<ceil_div_i(NPX * 16, 256), 256, 0, stream>>>(
        ws + o_c3, ws + o_st3, g3w, g3b, B, H, W);

    // conv4 16->8 3x3 + GN(4) + ReLU      (LDS: 3*18*20*4 = 4320 B)
    conv_wmma_kernel<16, 8, 16, 3, 1><<<ntiles, 32, 3 * 18 * 20 * 4, stream>>>(
        ws + o_c3, ws + o_wt4, b4, ws + o_c4, H, W);
    gn_reduce_kernel<8, 4><<<B * 4, 256, 0, stream>>>(ws + o_c4, ws + o_st4, H, W);
    gn_apply_kernel<8, 4, true><<<ceil_div_i(NPX * 8, 256), 256, 0, stream>>>(
        ws + o_c4, ws + o_st4, g4w, g4b, B, H, W);

    // conv5 8->2 3x3 (no GN)              (LDS: 3*18*12*4 = 2592 B)
    conv_wmma_kernel<8, 2, 16, 3, 1><<<ntiles, 32, 3 * 18 * 12 * 4, stream>>>(
        ws + o_c4, ws + o_wt5, b5, ws + o_c5, H, W);

    // guide + joint bilateral
    guide_kernel<<<ceil_div_i(NPX, 256), 256, 0, stream>>>(gt, ws + o_guide, B, H, W);
    bilateral_kernel<<<ceil_div_i(NPX, 256), 256, 0, stream>>>(
        ws + o_c5, ws + o_guide, ws + o_fb, B, H, W);

    // global min/max per channel, then compose to RGB
    mm_init_kernel<<<1, 1, 0, stream>>>(mm);
    mm_reduce_kernel<<<512, 256, 0, stream>>>(ws + o_fb, mm, NPX);
    compose_kernel<<<ceil_div_i(NPX, 256), 256, 0, stream>>>(
        x, ws + o_fb, mm, (float*)d_out, B, H, W);
}